// TorchDispatchModule_28879360099093
// MI455X (gfx1250) — compile-verified
//
#include <hip/hip_runtime.h>

#define C_   8
#define EPC_ 4
#define E_   32
#define K_   8
#define S_   256
#define H_   2048
#define M_   2048
#define ML_  8
#define SK_  (S_ * K_)   // 2048 picks per chip

typedef float v4f __attribute__((ext_vector_type(4)));

// ---------------------------------------------------------------------------
// Kernel 1: initialize inverse map (expert-slot -> packed source) to -1.
// inv has E_*M_ = 65536 int32 entries living in d_ws.
// ---------------------------------------------------------------------------
__global__ __launch_bounds__(256) void init_inv(int4* __restrict__ inv4) {
  const int gid = blockIdx.x * 256 + threadIdx.x;
  if (gid < (E_ * M_) / 4) inv4[gid] = make_int4(-1, -1, -1, -1);
}

// ---------------------------------------------------------------------------
// Kernel 2: one block per source chip. Reproduces the sequential
// "rank = #earlier picks of same expert on same chip" bookkeeping.
// 8 waves x 32 lanes: lane e owns expert e; per-wave chunk counts, then a
// cross-wave exclusive prefix, then an emit pass writes
//   inv[e*M + (base + rank)] = c*2048 + p   (dropped if dst out of [0,M)).
// ---------------------------------------------------------------------------
__global__ __launch_bounds__(256) void build_map(const int* __restrict__ indices,
                                                 const int* __restrict__ expert_offsets,
                                                 int* __restrict__ inv) {
  __shared__ int sidx[SK_];
  __shared__ int counts[8][E_];

  const int c   = blockIdx.x;
  const int tid = threadIdx.x;

  for (int i = tid; i < SK_; i += 256) sidx[i] = indices[c * SK_ + i];
  __syncthreads();

  const int w  = tid >> 5;   // wave id within block (0..7)
  const int e  = tid & 31;   // expert owned by this lane
  const int p0 = w * 256;

  int cnt = 0;
  for (int p = p0; p < p0 + 256; ++p) cnt += (sidx[p] == e) ? 1 : 0;
  counts[w][e] = cnt;
  __syncthreads();

  int rank = 0;
  for (int ww = 0; ww < w; ++ww) rank += counts[ww][e];

  const int base = expert_offsets[c * E_ + e];
  for (int p = p0; p < p0 + 256; ++p) {
    if (sidx[p] == e) {
      const int dst = base + rank;
      ++rank;
      if (dst >= 0 && dst < M_) inv[e * M_ + dst] = c * SK_ + p;  // mode='drop'
    }
  }
}

// ---------------------------------------------------------------------------
// Kernel 3: one block per output row (E*M = 65536 rows of H=2048 f32 = 8KB).
//  - unmapped row: nontemporal zero fill (512MB total store stream).
//  - mapped row:   CDNA5 async DMA copy  global -> LDS -> global
//                  (global_load_async_to_lds_b128 / global_store_async_from_lds_b128,
//                   tracked by ASYNCcnt, workgroup barrier between phases).
//  - threads 0..7 write the 8-int meta row (bf16 RNE bits of the gate weight).
// Uses 8KB of *dynamic* LDS (offset 0) addressed directly by byte offset.
// ---------------------------------------------------------------------------
__global__ __launch_bounds__(256) void write_rows(const float* __restrict__ x,
                                                  const float* __restrict__ weights,
                                                  const int* __restrict__ inv,
                                                  float* __restrict__ buf,
                                                  int* __restrict__ meta) {
  const int r   = blockIdx.x;        // row id: e = r>>11, m = r&2047
  const int t   = threadIdx.x;
  const int src = inv[r];            // block-uniform

  // ---- meta row -----------------------------------------------------------
  if (t < ML_) {
    int v = -1;
    if (src >= 0) {
      const int c = src >> 11;
      const int p = src & (SK_ - 1);
      if (t == 0)      v = c;                 // chip
      else if (t == 1) v = p >> 3;            // token
      else if (t == 2) v = p & (K_ - 1);      // topk
      else if (t == 3) v = r >> 11;           // expert idx
      else if (t == 4) {                      // bf16(weight) bits, sext int16
        const unsigned u = __float_as_uint(weights[c * SK_ + p]);
        const unsigned b = (u + 0x7FFFu + ((u >> 16) & 1u)) >> 16;  // RNE
        v = (int)(short)(unsigned short)b;
      } else v = 0;
    }
    meta[(size_t)r * ML_ + t] = v;
  }

  float* drow = buf + (size_t)r * H_;

  if (src < 0) {
    // ---- zero row: streaming stores (dominant 512MB stream) ---------------
    const v4f z = {0.f, 0.f, 0.f, 0.f};
    v4f* d4 = (v4f*)drow;
    __builtin_nontemporal_store(z, d4 + t);
    __builtin_nontemporal_store(z, d4 + t + 256);
  } else {
    // ---- data row: async DMA global -> LDS -> global ----------------------
    const int c     = src >> 11;
    const int token = (src & (SK_ - 1)) >> 3;
    const float* srow = x + ((size_t)c * S_ + token) * H_;

    const unsigned loff0 = (unsigned)(t * 16);     // LDS byte offsets (dyn LDS @ 0)
    const unsigned loff1 = loff0 + 4096u;
    unsigned long long ga0 = (unsigned long long)(uintptr_t)srow + (unsigned)(t * 16);
    unsigned long long ga1 = ga0 + 4096ull;

    asm volatile("global_load_async_to_lds_b128 %0, %1, off"
                 :: "v"(loff0), "v"(ga0) : "memory");
    asm volatile("global_load_async_to_lds_b128 %0, %1, off"
                 :: "v"(loff1), "v"(ga1) : "memory");
    asm volatile("s_wait_asynccnt 0" ::: "memory");
    __syncthreads();  // LDS writes of all waves visible before store phase

    unsigned long long gd0 = (unsigned long long)(uintptr_t)drow + (unsigned)(t * 16);
    unsigned long long gd1 = gd0 + 4096ull;
    asm volatile("global_store_async_from_lds_b128 %0, %1, off"
                 :: "v"(gd0), "v"(loff0) : "memory");
    asm volatile("global_store_async_from_lds_b128 %0, %1, off"
                 :: "v"(gd1), "v"(loff1) : "memory");
    asm volatile("s_wait_asynccnt 0" ::: "memory");
  }
}

// ---------------------------------------------------------------------------
extern "C" void kernel_launch(void* const* d_in, const int* in_sizes, int n_in,
                              void* d_out, int out_size, void* d_ws, size_t ws_size,
                              hipStream_t stream) {
  const float* x              = (const float*)d_in[0];  // (C,S,H) f32
  const float* weights        = (const float*)d_in[1];  // (C,S,K) f32
  const int*   indices        = (const int*)d_in[2];    // (C,S,K) i32
  const int*   expert_offsets = (const int*)d_in[3];    // (C,E)   i32

  float* buf  = (float*)d_out;                                    // (1,C,EPC,M,H)
  int*   meta = (int*)((float*)d_out + (size_t)E_ * M_ * H_);     // (1,C,EPC,M,ML)
  int*   inv  = (int*)d_ws;                                       // E*M int32

  init_inv<<<(E_ * M_) / 4 / 256, 256, 0, stream>>>((int4*)inv);
  build_map<<<C_, 256, 0, stream>>>(indices, expert_offsets, inv);
  write_rows<<<E_ * M_, 256, 8192, stream>>>(x, weights, inv, buf, meta);
}